// GraphGeneratorFixed_65910568124749
// MI455X (gfx1250) — compile-verified
//
#include <hip/hip_runtime.h>

// ---------------------------------------------------------------------------
// Fixed-capacity cutoff neighbor list, 64 systems x 1000 atoms, cutoff 5A.
//
// Distance tiles via V_WMMA_F32_16X16X4_F32:
//   A row m  = ( x_m,  y_m,  z_m, 1 )
//   B col n  = (-2x_n,-2y_n,-2z_n, |x_n|^2 )
//   C[m][n]  = |x_m|^2
//   => D[m][n] = |x_m|^2 + |x_n|^2 - 2 x_m.x_n
//
// Coords staged into LDS with the Tensor Data Mover (TENSORcnt +
// s_wait_tensorcnt), then expanded in LDS to WMMA-ready 16B fragments so each
// lane's A/B fragment is a single aligned ds_load_b64. Interior tiles ballot
// on the bare float compare (single v_cmp into SGPR); only the diagonal and
// last tiles evaluate the full (j>i, j<NAT, i<NAT) predicate. Ordered
// wave32 ballot/mbcnt compaction reproduces the reference's flat (s,i,j)
// cumsum order exactly.
// ---------------------------------------------------------------------------

#define NSYS   64
#define NAT    1000
#define NPAD   1008          // 63 tiles of 16
#define NTILE  63
#define MAXP   2200000
#define CUT2   25.0f
#define PADVAL 64000         // padding_value = total atom count

#define RAW_OFF 0            // 3000 raw coord floats  (12000 B)
#define A_OFF   12000        // 1008 float4 (x,y,z,1)   (16128 B)
#define B_OFF   28128        // 1008 float4 (-2x,-2y,-2z,sq)
#define LDS_BYTES 44256

typedef __attribute__((ext_vector_type(2))) float    v2f;
typedef __attribute__((ext_vector_type(8))) float    v8f;
typedef __attribute__((ext_vector_type(4))) unsigned v4u;
typedef __attribute__((ext_vector_type(4))) int      v4i;
typedef __attribute__((ext_vector_type(8))) int      v8i;

__device__ __forceinline__ unsigned ballot32(bool p) {
  return __builtin_amdgcn_ballot_w32(p);
}

// --------------------------------------------------------------- TDM staging
// DMA one 1-D run of f32 into LDS via the Tensor Data Mover, wait TENSORcnt.
__device__ __forceinline__ void tdm_copy_1d(unsigned lds_off, const void* gsrc,
                                            int nelem) {
#if __has_builtin(__builtin_amdgcn_tensor_load_to_lds)
  unsigned long long ga = (unsigned long long)(__SIZE_TYPE__)gsrc;
  v4u g0;
  g0[0] = 1u;                                       // count=1 valid descriptor
  g0[1] = lds_off;                                  // LDS byte address
  g0[2] = (unsigned)(ga & 0xFFFFFFFFu);             // global_addr[31:0]
  g0[3] = (unsigned)((ga >> 32) & 0x01FFFFFFu)      // global_addr[56:32]
        | 0x80000000u;                              // type = 2 ("image")
  v8i g1;
  g1[0] = (2 << 16);                                // data_size code 2 = 4 B
  g1[1] = (nelem & 0xFFFF) << 16;                   // tensor_dim0[15:0]
  g1[2] = ((nelem >> 16) & 0xFFFF) | (1 << 16);     // dim0[31:16] | dim1=1
  g1[3] = (nelem & 0xFFFF) << 16;                   // dim1 hi=0 | tile_dim0
  g1[4] = 1;                                        // tile_dim1=1 | tile_dim2=0
  g1[5] = nelem;                                    // tensor_dim0_stride[31:0]
  g1[6] = 0;
  g1[7] = 0;
  v4i gz = {0, 0, 0, 0};
#if __clang_major__ >= 23
  v8i gz8 = {0, 0, 0, 0, 0, 0, 0, 0};
  __builtin_amdgcn_tensor_load_to_lds(g0, g1, gz, gz, gz8, 0);
#else
  __builtin_amdgcn_tensor_load_to_lds(g0, g1, gz, gz, 0);
#endif
  __builtin_amdgcn_s_wait_tensorcnt(0);
#else
  (void)lds_off; (void)gsrc; (void)nelem;
#endif
}

// Stage coords of system s into LDS and expand to packed A/B fragments.
__device__ __forceinline__ void stage_system(const float* __restrict__ coords,
                                             int s, char* smem,
                                             unsigned lds_base) {
  float* raw = (float*)(smem + RAW_OFF);
#if __has_builtin(__builtin_amdgcn_tensor_load_to_lds)
  if ((threadIdx.x >> 5) == 0) {                    // one wave issues the DMA
    tdm_copy_1d(lds_base + RAW_OFF, coords + s * 3 * NAT, 3 * NAT);
  }
#else
  for (int e = threadIdx.x; e < 3 * NAT; e += blockDim.x)
    raw[e] = coords[s * 3 * NAT + e];
#endif
  __syncthreads();
  float* af = (float*)(smem + A_OFF);
  float* bf = (float*)(smem + B_OFF);
  for (int a = threadIdx.x; a < NPAD; a += blockDim.x) {
    float x = 0.f, y = 0.f, z = 0.f, w = 0.f;
    if (a < NAT) {
      x = raw[3 * a + 0];
      y = raw[3 * a + 1];
      z = raw[3 * a + 2];
      w = 1.0f;
    }
    af[4 * a + 0] = x;
    af[4 * a + 1] = y;
    af[4 * a + 2] = z;
    af[4 * a + 3] = w;
    bf[4 * a + 0] = -2.0f * x;
    bf[4 * a + 1] = -2.0f * y;
    bf[4 * a + 2] = -2.0f * z;
    bf[4 * a + 3] = x * x + y * y + z * z;   // |x|^2 (0 for padding)
  }
  __syncthreads();
}

// ---------------------------------------------------------------- output defaults
__global__ void k_init(float* __restrict__ out) {
  int g = blockIdx.x * blockDim.x + threadIdx.x;
  if (g < 4 * MAXP) {
    out[g] = (float)PADVAL;            // edge_src + edge_dst regions
  } else if (g < 6 * MAXP) {
    out[g] = CUT2;                     // d12 region defaults to cutoff^2
  }
}

// ---------------------------------------------------------------- count pass
__global__ void k_count(const float* __restrict__ coords,
                        int* __restrict__ rowcnt) {
  extern __shared__ char smem[];
  unsigned lds_base = __builtin_amdgcn_groupstaticsize();  // dynamic LDS start

  int s  = blockIdx.x >> 3;              // 8 blocks per system
  int ti = (blockIdx.x & 7) * 8 + (threadIdx.x >> 5);
  int lane = threadIdx.x & 31;
  int half = lane >> 4;
  int lp   = lane & 15;

  stage_system(coords, s, smem, lds_base);

  if (ti >= NTILE) return;               // wave-uniform
  int i0 = ti * 16;
  unsigned hmask = 0xFFFFu << (16 * half);

  const v2f* avec = (const v2f*)(smem + A_OFF);
  const v2f* bvec = (const v2f*)(smem + B_OFF);
  const float* bf = (const float*)(smem + B_OFF);

  v2f a = avec[2 * (i0 + lp) + half];    // (x,y) or (z,1) by lane half

  v8f c;                                 // |x_row|^2 per VGPR row
#pragma unroll
  for (int r = 0; r < 8; ++r) c[r] = bf[4 * (i0 + 8 * half + r) + 3];

  int cnt[8];
#pragma unroll
  for (int r = 0; r < 8; ++r) cnt[r] = 0;

  // diagonal tile: full predicate
  {
    int tj = ti;
    int j  = tj * 16 + lp;
    v2f b = bvec[2 * j + half];
    v8f d = __builtin_amdgcn_wmma_f32_16x16x4_f32(
        false, a, false, b, (short)0, c, false, false);
#pragma unroll
    for (int r = 0; r < 8; ++r) {
      int i = i0 + 8 * half + r;
      bool pred = (d[r] < CUT2) && (j > i) && (j < NAT) && (i < NAT);
      cnt[r] += __builtin_popcount(ballot32(pred) & hmask);
    }
  }
  // interior tiles: j>i, j<NAT, i<NAT all guaranteed
  for (int tj = ti + 1; tj < NTILE - 1; ++tj) {
    v2f b = bvec[2 * (tj * 16 + lp) + half];
    v8f d = __builtin_amdgcn_wmma_f32_16x16x4_f32(
        false, a, false, b, (short)0, c, false, false);
#pragma unroll
    for (int r = 0; r < 8; ++r)
      cnt[r] += __builtin_popcount(ballot32(d[r] < CUT2) & hmask);
  }
  // last tile: needs j<NAT
  if (ti < NTILE - 1) {
    int tj = NTILE - 1;
    int j  = tj * 16 + lp;
    v2f b = bvec[2 * j + half];
    v8f d = __builtin_amdgcn_wmma_f32_16x16x4_f32(
        false, a, false, b, (short)0, c, false, false);
#pragma unroll
    for (int r = 0; r < 8; ++r) {
      bool pred = (d[r] < CUT2) && (j < NAT);
      cnt[r] += __builtin_popcount(ballot32(pred) & hmask);
    }
  }

  if (lp == 0) {   // lane 0 writes rows i0..i0+7, lane 16 writes i0+8..i0+15
#pragma unroll
    for (int r = 0; r < 8; ++r)
      rowcnt[s * NPAD + i0 + 8 * half + r] = cnt[r];
  }
}

// ---------------------------------------------------------------- exclusive scan
__global__ void k_scan(const int* __restrict__ rowcnt,
                       int* __restrict__ rowoff,
                       float* __restrict__ out_npairs) {
  __shared__ int sh[256];
  const int NROWS = NSYS * NPAD;       // 64512 = 252 * 256
  int tid = threadIdx.x;
  int carry = 0;
  for (int base = 0; base < NROWS; base += 256) {
    int v = rowcnt[base + tid];
    sh[tid] = v;
    __syncthreads();
    for (int off = 1; off < 256; off <<= 1) {
      int t = (tid >= off) ? sh[tid - off] : 0;
      __syncthreads();
      if (tid >= off) sh[tid] += t;
      __syncthreads();
    }
    int incl  = sh[tid];
    int total = sh[255];
    __syncthreads();
    rowoff[base + tid] = carry + incl - v;
    carry += total;
  }
  if (tid == 0) *out_npairs = (float)carry;
}

// ---------------------------------------------------------------- write pass
__global__ void k_write(const float* __restrict__ coords,
                        const int* __restrict__ rowoff,
                        float* __restrict__ out) {
  extern __shared__ char smem[];
  unsigned lds_base = __builtin_amdgcn_groupstaticsize();

  int s  = blockIdx.x >> 3;
  int ti = (blockIdx.x & 7) * 8 + (threadIdx.x >> 5);
  int lane = threadIdx.x & 31;
  int half = lane >> 4;
  int lp   = lane & 15;

  stage_system(coords, s, smem, lds_base);

  if (ti >= NTILE) return;               // wave-uniform
  int i0 = ti * 16;
  unsigned hmask = 0xFFFFu << (16 * half);

  const v2f* avec = (const v2f*)(smem + A_OFF);
  const v2f* bvec = (const v2f*)(smem + B_OFF);
  const float* bf = (const float*)(smem + B_OFF);

  v2f a = avec[2 * (i0 + lp) + half];

  v8f c;
  int off[8];
#pragma unroll
  for (int r = 0; r < 8; ++r) {
    int irow = i0 + 8 * half + r;
    c[r]   = bf[4 * irow + 3];
    off[r] = rowoff[s * NPAD + irow];
  }

  // --- diagonal tile: full predicate
  {
    int tj = ti;
    int j  = tj * 16 + lp;
    v2f b = bvec[2 * j + half];
    v8f d = __builtin_amdgcn_wmma_f32_16x16x4_f32(
        false, a, false, b, (short)0, c, false, false);
#pragma unroll
    for (int r = 0; r < 8; ++r) {
      int i = i0 + 8 * half + r;
      bool pred = (d[r] < CUT2) && (j > i) && (j < NAT) && (i < NAT);
      unsigned hm = ballot32(pred) & hmask;
      if (pred) {
        int pos = off[r] + (int)__builtin_amdgcn_mbcnt_lo(hm, 0);
        if (pos < MAXP) {
          float fsrc = (float)(s * NAT + i);
          float fdst = (float)(s * NAT + j);
          float dd   = d[r];
          out[0 * MAXP + pos] = fsrc;  out[1 * MAXP + pos] = fdst;
          out[2 * MAXP + pos] = fdst;  out[3 * MAXP + pos] = fsrc;
          out[4 * MAXP + pos] = dd;    out[5 * MAXP + pos] = dd;
        }
      }
      off[r] += __builtin_popcount(hm);
    }
  }
  // --- interior tiles
  for (int tj = ti + 1; tj < NTILE - 1; ++tj) {
    int j = tj * 16 + lp;
    v2f b = bvec[2 * j + half];
    v8f d = __builtin_amdgcn_wmma_f32_16x16x4_f32(
        false, a, false, b, (short)0, c, false, false);
#pragma unroll
    for (int r = 0; r < 8; ++r) {
      bool pred = (d[r] < CUT2);
      unsigned hm = ballot32(pred) & hmask;
      if (pred) {
        int pos = off[r] + (int)__builtin_amdgcn_mbcnt_lo(hm, 0);
        if (pos < MAXP) {
          int i = i0 + 8 * half + r;
          float fsrc = (float)(s * NAT + i);
          float fdst = (float)(s * NAT + j);
          float dd   = d[r];
          out[0 * MAXP + pos] = fsrc;  out[1 * MAXP + pos] = fdst;
          out[2 * MAXP + pos] = fdst;  out[3 * MAXP + pos] = fsrc;
          out[4 * MAXP + pos] = dd;    out[5 * MAXP + pos] = dd;
        }
      }
      off[r] += __builtin_popcount(hm);
    }
  }
  // --- last tile: needs j<NAT
  if (ti < NTILE - 1) {
    int tj = NTILE - 1;
    int j  = tj * 16 + lp;
    v2f b = bvec[2 * j + half];
    v8f d = __builtin_amdgcn_wmma_f32_16x16x4_f32(
        false, a, false, b, (short)0, c, false, false);
#pragma unroll
    for (int r = 0; r < 8; ++r) {
      bool pred = (d[r] < CUT2) && (j < NAT);
      unsigned hm = ballot32(pred) & hmask;
      if (pred) {
        int pos = off[r] + (int)__builtin_amdgcn_mbcnt_lo(hm, 0);
        if (pos < MAXP) {
          int i = i0 + 8 * half + r;
          float fsrc = (float)(s * NAT + i);
          float fdst = (float)(s * NAT + j);
          float dd   = d[r];
          out[0 * MAXP + pos] = fsrc;  out[1 * MAXP + pos] = fdst;
          out[2 * MAXP + pos] = fdst;  out[3 * MAXP + pos] = fsrc;
          out[4 * MAXP + pos] = dd;    out[5 * MAXP + pos] = dd;
        }
      }
      off[r] += __builtin_popcount(hm);
    }
  }
}

// ---------------------------------------------------------------------------
extern "C" void kernel_launch(void* const* d_in, const int* in_sizes, int n_in,
                              void* d_out, int out_size, void* d_ws, size_t ws_size,
                              hipStream_t stream) {
  (void)in_sizes; (void)n_in; (void)out_size; (void)ws_size;
  const float* coords = (const float*)d_in[0];   // [64000, 3] float32
  // d_in[1] = isys, d_in[2] = natoms : shapes are static, unused.
  float* out = (float*)d_out;                    // 6*MAXP + 1 floats

  int* rowcnt = (int*)d_ws;                      // 64512 ints
  int* rowoff = rowcnt + NSYS * NPAD;            // 64512 ints

  int nblk = NSYS * 8;                           // 8 tile-row waves per block

  k_init <<<(6 * MAXP + 255) / 256, 256, 0, stream>>>(out);
  k_count<<<nblk, 256, LDS_BYTES, stream>>>(coords, rowcnt);
  k_scan <<<1, 256, 0, stream>>>(rowcnt, rowoff, out + 6 * MAXP);
  k_write<<<nblk, 256, LDS_BYTES, stream>>>(coords, rowoff, out);
}